// TensorProductWeightsBlock_1288490189568
// MI455X (gfx1250) — compile-verified
//
#include <hip/hip_runtime.h>

typedef __attribute__((ext_vector_type(2))) float v2f;
typedef __attribute__((ext_vector_type(8))) float v8f;

#define N_EDGES_C 800000
#define N_ELEM_C  10
#define F_IN_C    128
#define NPAIR_C   100        // N_ELEM * N_ELEM
#define XSTRIDE   132        // 128 + 4 pad: A-fragment column reads hit 16 distinct LDS banks
#define EPB       256        // edges per block (8 waves x 32)
#define WDP_PAIRS 50         // 100 K-rows -> 50 pairs
#define W2P_PAIRS 64         // 128 K-rows -> 64 pairs

// ---------------------------------------------------------------------------
// Prep kernel: build K-pair-packed B matrices so each WMMA B-fragment is ONE
// global_load_b64 per lane (pair p holds rows k=2p and k=2p+1 of the matrix).
//   wdp[p][n] = { W1[.,.,n,n] at c=2p , c=2p+1 }   ([50][128] float2)
//   w2p[p][n] = { W2[2p][n]  , W2[2p+1][n] }       ([64][128] float2)
// ---------------------------------------------------------------------------
__global__ void pack_weights_kernel(const float* __restrict__ w1,
                                    const float* __restrict__ w2,
                                    v2f* __restrict__ wdp,
                                    v2f* __restrict__ w2p) {
    int idx = blockIdx.x * 256 + threadIdx.x;        // over max(6400, 8192)
    int n  = idx & (F_IN_C - 1);
    int p  = idx >> 7;
    if (idx < WDP_PAIRS * F_IN_C) {
        v2f v;
        v.x = w1[((size_t)(2 * p    ) * F_IN_C + n) * F_IN_C + n];
        v.y = w1[((size_t)(2 * p + 1) * F_IN_C + n) * F_IN_C + n];
        wdp[idx] = v;
    }
    if (idx < W2P_PAIRS * F_IN_C) {
        v2f v;
        v.x = w2[(2 * p    ) * F_IN_C + n];
        v.y = w2[(2 * p + 1) * F_IN_C + n];
        w2p[idx] = v;
    }
}

// ---------------------------------------------------------------------------
// Fused kernel: 8 waves/block, 32 edges/wave (two 16-row M-tiles sharing every
// B-fragment load), 256 edges/block, 3125 blocks.
//   t  = (a_s outer a_r) [32x100] @ Wd [100x128]    (25 k-steps, 2 wmma each)
//   x  = edge_feats * t                             (C-layout modulate -> LDS)
//   out= x [32x128] @ W2 [128x128]                  (32 k-steps, 2 wmma each)
// ---------------------------------------------------------------------------
__global__ void __launch_bounds__(256)
fused_edge_kernel(const float* __restrict__ node_attrs,     // [N,10]
                  const float* __restrict__ edge_feats,     // [E,128]
                  const long long* __restrict__ edge_index, // [2,E] int64
                  const v2f* __restrict__ wdp,              // [50][128]
                  const v2f* __restrict__ w2p,              // [64][128]
                  float* __restrict__ out)                  // [E,128]
{
    extern __shared__ float smem[];
    float* as_s = smem;                          // [256][10]  sender attrs
    float* ar_s = smem + EPB * N_ELEM_C;         // [256][10]  receiver attrs
    float* x_s  = smem + 2 * EPB * N_ELEM_C;     // [256][XSTRIDE] modulated feats

    const int tid  = threadIdx.x;
    const int wave = tid >> 5;
    const int lane = tid & 31;
    const long long edge0 = (long long)blockIdx.x * EPB;

    // ---- cooperative gather of node attrs for this block's 256 edges ----
    for (int q = tid; q < EPB * N_ELEM_C; q += 256) {
        int e = q / N_ELEM_C;
        int i = q - e * N_ELEM_C;
        long long ge   = edge0 + e;
        long long srow = edge_index[ge];                        // sender
        long long rrow = edge_index[(long long)N_EDGES_C + ge]; // receiver
        as_s[q] = node_attrs[srow * N_ELEM_C + i];
        ar_s[q] = node_attrs[rrow * N_ELEM_C + i];
    }
    __syncthreads();

    const int m16   = lane & 15;            // row within a 16-edge M-tile
    const int kq    = lane >> 4;            // K quadrant select (0 / 1)
    const int khalf = kq << 1;              // K sub-offset: 0 or 2
    const int mbase = kq << 3;              // C-layout row base: 0 or 8
    const int erow0 = wave * 32 + m16;      // M-tile 0 row within block
    const int erow1 = erow0 + 16;           // M-tile 1 row within block
    const long long gedge0 = edge0 + wave * 32;

    // ---- A-fragments of p = a_s (outer) a_r for both M-tiles (reg-resident)
    v2f afr0[25], afr1[25];
    {
        const float* ase0 = as_s + erow0 * N_ELEM_C;
        const float* are0 = ar_s + erow0 * N_ELEM_C;
        const float* ase1 = as_s + erow1 * N_ELEM_C;
        const float* are1 = ar_s + erow1 * N_ELEM_C;
#pragma unroll
        for (int t = 0; t < 25; ++t) {
            int ka = t * 4 + khalf;         // pair index in [0,100)
            int kb = ka + 1;
            int ia = ka / N_ELEM_C, ja = ka % N_ELEM_C;
            int ib = kb / N_ELEM_C, jb = kb % N_ELEM_C;
            afr0[t].x = ase0[ia] * are0[ja];
            afr0[t].y = ase0[ib] * are0[jb];
            afr1[t].x = ase1[ia] * are1[ja];
            afr1[t].y = ase1[ib] * are1[jb];
        }
    }

    // ---- Phase A: t = p @ Wd; modulate by edge_feats; stage into LDS ----
    for (int ntile = 0; ntile < 8; ++ntile) {
        const int ncol = ntile * 16 + m16;
        v8f acc0 = {};
        v8f acc1 = {};
#pragma unroll
        for (int t = 0; t < 25; ++t) {
            v2f b = wdp[(t * 2 + kq) * F_IN_C + ncol];   // one b64, shared
            acc0 = __builtin_amdgcn_wmma_f32_16x16x4_f32(
                       false, afr0[t], false, b, (short)0, acc0, false, false);
            acc1 = __builtin_amdgcn_wmma_f32_16x16x4_f32(
                       false, afr1[t], false, b, (short)0, acc1, false, false);
        }
#pragma unroll
        for (int v = 0; v < 8; ++v) {
            int m = v + mbase;
            float ef0 = edge_feats[(gedge0 + m     ) * F_IN_C + ncol];
            float ef1 = edge_feats[(gedge0 + m + 16) * F_IN_C + ncol];
            x_s[(wave * 32 + m     ) * XSTRIDE + ncol] = acc0[v] * ef0;
            x_s[(wave * 32 + m + 16) * XSTRIDE + ncol] = acc1[v] * ef1;
        }
    }
    // x_s written/read only by the owning wave; LDS ops are in-order per wave.

    // ---- Phase B: out = x @ W2 ----
    for (int ntile = 0; ntile < 8; ++ntile) {
        const int ncol = ntile * 16 + m16;
        v8f acc0 = {};
        v8f acc1 = {};
#pragma unroll 8
        for (int t = 0; t < 32; ++t) {
            int ka = t * 4 + khalf;
            v2f a0 = *(const v2f*)(x_s + erow0 * XSTRIDE + ka);  // ds_load_b64
            v2f a1 = *(const v2f*)(x_s + erow1 * XSTRIDE + ka);  // ds_load_b64
            v2f b  = w2p[(t * 2 + kq) * F_IN_C + ncol];          // one b64, shared
            acc0 = __builtin_amdgcn_wmma_f32_16x16x4_f32(
                       false, a0, false, b, (short)0, acc0, false, false);
            acc1 = __builtin_amdgcn_wmma_f32_16x16x4_f32(
                       false, a1, false, b, (short)0, acc1, false, false);
        }
#pragma unroll
        for (int v = 0; v < 8; ++v) {
            int m = v + mbase;
            out[(gedge0 + m     ) * F_IN_C + ncol] = acc0[v];
            out[(gedge0 + m + 16) * F_IN_C + ncol] = acc1[v];
        }
    }
}

// ---------------------------------------------------------------------------
extern "C" void kernel_launch(void* const* d_in, const int* in_sizes, int n_in,
                              void* d_out, int out_size, void* d_ws, size_t ws_size,
                              hipStream_t stream) {
    const float*     node_attrs = (const float*)d_in[0];
    const float*     edge_feats = (const float*)d_in[1];
    const long long* edge_index = (const long long*)d_in[2];
    const float*     weights1   = (const float*)d_in[3];
    const float*     weights2   = (const float*)d_in[4];
    float* out = (float*)d_out;

    v2f* wdp = (v2f*)d_ws;                              // 50*128*8  = 51200 B
    v2f* w2p = (v2f*)((char*)d_ws + WDP_PAIRS * F_IN_C * sizeof(v2f)); // +51200 B

    // pack diag(W1) and W2 into K-pair layouts every call (cheap, deterministic)
    pack_weights_kernel<<<(W2P_PAIRS * F_IN_C + 255) / 256, 256, 0, stream>>>(
        weights1, weights2, wdp, w2p);

    // dynamic LDS: 2*256*10 attrs + 256*132 staged x  (floats) = 155648 B
    size_t shmem = (size_t)(2 * EPB * N_ELEM_C + EPB * XSTRIDE) * sizeof(float);
    fused_edge_kernel<<<N_EDGES_C / EPB, 256, shmem, stream>>>(
        node_attrs, edge_feats, edge_index, wdp, w2p, out);

    (void)in_sizes; (void)n_in; (void)out_size; (void)ws_size;
}